// ModelNew_70918499991666
// MI455X (gfx1250) — compile-verified
//
#include <hip/hip_runtime.h>

// Gated delta-rule linear attention (KDA), chunked-parallel formulation for
// CDNA5 / gfx1250.
//  - All matmuls via V_WMMA_F32_16X16X4_F32 (full f32, matches f32 reference).
//  - Chunk staging via GLOBAL_LOAD_ASYNC_TO_LDS_B128 (ASYNCcnt) with LDS
//    ping-pong double buffering; the chunk loop is unrolled by 2 with
//    statically swapped buffers so all LDS pointers keep their address-space
//    provenance (pure DS instructions, no generic FLAT fallback).
//  - grid = B*H*2 = 128 blocks x 128 threads (4 wave32 waves); per-wave
//    16-column V slab of the LDS-resident state S[128][64].

#define B_DIM 4
#define T_DIM 2048
#define H_DIM 16
#define K_DIM 128
#define V_DIM 128
#define CHUNK 16
#define NCHUNK (T_DIM / CHUNK)
#define VSPLIT 2
#define VSEG   (V_DIM / VSPLIT)      // 64 v-columns per block
#define LROW   132                    // row stride (floats) for [CHUNK][K] buffers
#define VROW   72                     // row stride for [CHUNK][VSEG] buffers
#define SROW   72                     // row stride for state [K][VSEG]
#define NTHREADS 128
#define SCALE 0.08838834764831845f    // K^-0.5

#if defined(__HIP_DEVICE_COMPILE__) && \
    __has_builtin(__builtin_amdgcn_global_load_async_to_lds_b128)
#define KDA_ASYNC 1
#else
#define KDA_ASYNC 0
#endif

typedef float v2f __attribute__((ext_vector_type(2)));
typedef float v8f __attribute__((ext_vector_type(8)));

#if KDA_ASYNC
// The builtin wants pointers to 16-byte int vectors: AS(1) src, AS(3) dst.
typedef int b128_t __attribute__((vector_size(4 * sizeof(int))));
typedef __attribute__((address_space(1))) b128_t* gptr128_t;
typedef __attribute__((address_space(3))) b128_t* lptr128_t;
#endif

__device__ __forceinline__ v8f wmma_k4(v2f a, v2f b, v8f c) {
    // D(16x16,f32) = A(16x4,f32) * B(4x16,f32) + C
    return __builtin_amdgcn_wmma_f32_16x16x4_f32(
        /*neg_a=*/false, a, /*neg_b=*/false, b,
        /*c_mod=*/(short)0, c, /*reuse_a=*/false, /*reuse_b=*/false);
}

// 16B global -> LDS copy: async DMA (ASYNCcnt) when available.
__device__ __forceinline__ void cp16(const float* gsrc, float* ldst) {
#if KDA_ASYNC
    __builtin_amdgcn_global_load_async_to_lds_b128(
        (gptr128_t)(gsrc), (lptr128_t)(ldst), /*imm offset=*/0, /*cpol=*/0);
#else
    *(float4*)ldst = *(const float4*)gsrc;
#endif
}

__device__ __forceinline__ void async_wait_all() {
#if KDA_ASYNC
    asm volatile("s_wait_asynccnt 0x0" ::: "memory");
#endif
}

// Stage one chunk (16 timesteps) of q,k,g (full K) and v (VSEG slab) into LDS.
__device__ __forceinline__ void load_chunk(const float* __restrict__ qg,
                                           const float* __restrict__ kg,
                                           const float* __restrict__ gg,
                                           const float* __restrict__ vg,
                                           float* qr, float* kr, float* gr, float* vr,
                                           int b, int h, int vbase, int t0, int tid)
{
    #pragma unroll
    for (int it = 0; it < 4; ++it) {
        int idx = tid + it * NTHREADS;          // 0..511
        int row = idx >> 5;
        int c4  = (idx & 31) << 2;
        size_t go = (((size_t)(b * T_DIM + t0 + row) * H_DIM) + h) * (size_t)K_DIM + c4;
        cp16(qg + go, qr + row * LROW + c4);
        cp16(kg + go, kr + row * LROW + c4);
        cp16(gg + go, gr + row * LROW + c4);
        if (t0 + CHUNK < T_DIM) {               // prefetch the chunk after next
            size_t gn = go + (size_t)CHUNK * H_DIM * K_DIM;
            __builtin_prefetch(qg + gn, 0, 1);  // global_prefetch_b8
            __builtin_prefetch(kg + gn, 0, 1);
            __builtin_prefetch(gg + gn, 0, 1);
        }
    }
    #pragma unroll
    for (int it = 0; it < 2; ++it) {
        int idx = tid + it * NTHREADS;          // 0..255
        int row = idx >> 4;
        int c4  = (idx & 15) << 2;
        size_t go = (((size_t)(b * T_DIM + t0 + row) * H_DIM) + h) * (size_t)V_DIM
                    + vbase + c4;
        cp16(vg + go, vr + row * VROW + c4);
        if (t0 + CHUNK < T_DIM)
            __builtin_prefetch(vg + go + (size_t)CHUNK * H_DIM * V_DIM, 0, 1);
    }
}

__global__ __launch_bounds__(NTHREADS, 1)
void kda_fwd_kernel(const float* __restrict__ qg,
                    const float* __restrict__ kg,
                    const float* __restrict__ vg,
                    const float* __restrict__ gg,
                    const float* __restrict__ betag,
                    float* __restrict__ outg)
{
    extern __shared__ float smem[];
    float* S    = smem;                        // [K_DIM][SROW] state
    float* qr0  = S   + K_DIM * SROW;          // raw q -> q~ in place (ping)
    float* qr1  = qr0 + CHUNK * LROW;          // (pong)
    float* kr0  = qr1 + CHUNK * LROW;          // raw k -> k~ in place
    float* kr1  = kr0 + CHUNK * LROW;
    float* gr0  = kr1 + CHUNK * LROW;          // raw g -> Lambda in place
    float* gr1  = gr0 + CHUNK * LROW;
    float* vr0  = gr1 + CHUNK * LROW;          // raw v slab
    float* vr1  = vr0 + CHUNK * VROW;
    float* kbh  = vr1 + CHUNK * VROW;          // beta * k / Lambda
    float* U    = kbh + CHUNK * LROW;          // W -> delta
    float* As   = U   + CHUNK * VROW;          // [16][20] strict-lower A
    float* Aq   = As  + 16 * 20;               // [16][20] inclusive-lower Aq
    float* lamC = Aq  + 16 * 20;               // [K_DIM]
    float* betaS= lamC + K_DIM;                // [CHUNK]

    const int tid  = threadIdx.x;
    const int lane = tid & 31;
    const int wv   = tid >> 5;                 // wave 0..3
    const int lo   = lane & 15;
    const int hi2  = (lane >> 4) << 1;         // 0 or 2 (A/B frag K select)
    const int hr   = (lane >> 4) << 3;         // 0 or 8 (C/D row select)

    const int bh    = blockIdx.x >> 1;
    const int vhalf = blockIdx.x & 1;
    const int b     = bh / H_DIM;
    const int h     = bh % H_DIM;
    const int vbase = vhalf * VSEG;
    const int vc    = wv * 16 + lo;            // local v column 0..63

    for (int i = tid; i < K_DIM * SROW; i += NTHREADS) S[i] = 0.f;

    // One chunk of work.  qc/kc/gc/vcur: this chunk's (already streamed)
    // buffers; qn/kn/gn/vn: buffers for the next chunk's async DMA.
    auto chunk_body = [&](float* qc, float* kc, float* gc, float* vcur,
                          float* qn, float* kn, float* gn, float* vn,
                          int t0) {
        if (tid < CHUNK)
            betaS[tid] = betag[(size_t)(b * T_DIM + t0 + tid) * H_DIM + h];

        async_wait_all();          // this wave's async loads for this chunk landed
        __syncthreads();           // everyone's landed -> raw buffers readable

        // ---------- gating transforms: one thread per k channel (in place) ----------
        {
            const int kk = tid;                // NTHREADS == K_DIM
            float run = 0.f;
            #pragma unroll
            for (int t = 0; t < CHUNK; ++t) {
                run += gc[t * LROW + kk];
                gc[t * LROW + kk] = __expf(run);     // Lambda_t[k]
            }
            lamC[kk] = gc[(CHUNK - 1) * LROW + kk];
            #pragma unroll
            for (int t = 0; t < CHUNK; ++t) {
                float lam = gc[t * LROW + kk];
                float kx  = kc[t * LROW + kk];
                float bt  = betaS[t];
                kc [t * LROW + kk] = kx * lam;        // k~
                kbh[t * LROW + kk] = bt * kx / lam;   // beta*k/Lambda
                qc [t * LROW + kk] *= SCALE * lam;    // q~
            }
        }
        __syncthreads();

        // kick off next chunk's DMA now, overlapping all remaining WMMA work
        if (t0 + CHUNK < T_DIM)
            load_chunk(qg, kg, gg, vg, qn, kn, gn, vn, b, h, vbase, t0 + CHUNK, tid);

        // ---------- step 1: W = K~ @ S0 (each wave: its 16-col slab) ----------
        v8f Wf = {0.f,0.f,0.f,0.f,0.f,0.f,0.f,0.f};
        #pragma unroll
        for (int kb = 0; kb < 8; ++kb) {
            #pragma unroll
            for (int s = 0; s < 4; ++s) {
                int kcc = kb * 16 + s * 4 + hi2;
                v2f a  = { kc[lo * LROW + kcc], kc[lo * LROW + kcc + 1] };
                v2f bm = { S[kcc * SROW + vc], S[(kcc + 1) * SROW + vc] };
                Wf = wmma_k4(a, bm, Wf);
            }
        }
        #pragma unroll
        for (int j = 0; j < 8; ++j)
            U[(j + hr) * VROW + vc] = Wf[j];

        // wave 0: A = tril_strict(K~ @ K^b^T);  wave 1: Aq = tril_incl(Q~ @ K^b^T)
        if (wv == 0) {
            v8f Af = {0.f,0.f,0.f,0.f,0.f,0.f,0.f,0.f};
            #pragma unroll
            for (int kb = 0; kb < 8; ++kb) {
                #pragma unroll
                for (int s = 0; s < 4; ++s) {
                    int kcc = kb * 16 + s * 4 + hi2;
                    v2f a  = { kc [lo * LROW + kcc], kc [lo * LROW + kcc + 1] };
                    v2f bm = { kbh[lo * LROW + kcc], kbh[lo * LROW + kcc + 1] };
                    Af = wmma_k4(a, bm, Af);
                }
            }
            #pragma unroll
            for (int j = 0; j < 8; ++j) {
                int tr = j + hr;
                As[tr * 20 + lo] = (lo < tr) ? Af[j] : 0.f;
            }
        } else if (wv == 1) {
            v8f Af = {0.f,0.f,0.f,0.f,0.f,0.f,0.f,0.f};
            #pragma unroll
            for (int kb = 0; kb < 8; ++kb) {
                #pragma unroll
                for (int s = 0; s < 4; ++s) {
                    int kcc = kb * 16 + s * 4 + hi2;
                    v2f a  = { qc [lo * LROW + kcc], qc [lo * LROW + kcc + 1] };
                    v2f bm = { kbh[lo * LROW + kcc], kbh[lo * LROW + kcc + 1] };
                    Af = wmma_k4(a, bm, Af);
                }
            }
            #pragma unroll
            for (int j = 0; j < 8; ++j) {
                int tr = j + hr;
                Aq[tr * 20 + lo] = (lo <= tr) ? Af[j] : 0.f;
            }
        }
        __syncthreads();

        // ---------- step 2: forward substitution: delta = (I + trilA)^-1 (v - W) ----------
        if (tid < VSEG) {
            const int vcol = tid;
            #pragma unroll
            for (int t = 0; t < CHUNK; ++t) {
                float d = vcur[t * VROW + vcol] - U[t * VROW + vcol];
                #pragma unroll
                for (int i = 0; i < CHUNK; ++i)
                    if (i < t) d -= As[t * 20 + i] * U[i * VROW + vcol];
                U[t * VROW + vcol] = d;     // U now holds delta_t
            }
        }
        __syncthreads();

        // ---------- step 3: O = Q~ @ S0 + Aq @ Delta ----------
        v8f Of = {0.f,0.f,0.f,0.f,0.f,0.f,0.f,0.f};
        #pragma unroll
        for (int kb = 0; kb < 8; ++kb) {
            #pragma unroll
            for (int s = 0; s < 4; ++s) {
                int kcc = kb * 16 + s * 4 + hi2;
                v2f a  = { qc[lo * LROW + kcc], qc[lo * LROW + kcc + 1] };
                v2f bm = { S[kcc * SROW + vc], S[(kcc + 1) * SROW + vc] };
                Of = wmma_k4(a, bm, Of);
            }
        }
        #pragma unroll
        for (int s = 0; s < 4; ++s) {
            int ic = s * 4 + hi2;
            v2f a  = { Aq[lo * 20 + ic], Aq[lo * 20 + ic + 1] };
            v2f bm = { U[ic * VROW + vc], U[(ic + 1) * VROW + vc] };
            Of = wmma_k4(a, bm, Of);
        }
        #pragma unroll
        for (int j = 0; j < 8; ++j) {
            int tr = j + hr;
            __builtin_nontemporal_store(
                Of[j],
                outg + (((size_t)(b * T_DIM + t0 + tr) * H_DIM) + h) * (size_t)V_DIM
                     + vbase + vc);
        }

        // ---------- step 4: S = diag(LamC)*S0 + (beta*k*LamC/Lam)^T @ Delta ----------
        #pragma unroll
        for (int kb = 0; kb < 8; ++kb) {
            v8f acc;
            #pragma unroll
            for (int j = 0; j < 8; ++j) {
                int kr = kb * 16 + j + hr;
                acc[j] = S[kr * SROW + vc] * lamC[kr];
            }
            float lck = lamC[kb * 16 + lo];
            #pragma unroll
            for (int s = 0; s < 4; ++s) {
                int ic = s * 4 + hi2;
                v2f a  = { kbh[ic * LROW + kb * 16 + lo] * lck,
                           kbh[(ic + 1) * LROW + kb * 16 + lo] * lck };
                v2f bm = { U[ic * VROW + vc], U[(ic + 1) * VROW + vc] };
                acc = wmma_k4(a, bm, acc);
            }
            #pragma unroll
            for (int j = 0; j < 8; ++j) {
                int kr = kb * 16 + j + hr;
                S[kr * SROW + vc] = acc[j];
            }
        }
        __syncthreads();   // protect derived buffers / U / As / Aq for next chunk
    };

    // prologue: stream chunk 0 into ping buffers
    load_chunk(qg, kg, gg, vg, qr0, kr0, gr0, vr0, b, h, vbase, 0, tid);

    // two chunks per iteration with statically swapped ping/pong buffers
    for (int ch2 = 0; ch2 < NCHUNK; ch2 += 2) {
        chunk_body(qr0, kr0, gr0, vr0, qr1, kr1, gr1, vr1, (ch2 + 0) * CHUNK);
        chunk_body(qr1, kr1, gr1, vr1, qr0, kr0, gr0, vr0, (ch2 + 1) * CHUNK);
    }
}

extern "C" void kernel_launch(void* const* d_in, const int* in_sizes, int n_in,
                              void* d_out, int out_size, void* d_ws, size_t ws_size,
                              hipStream_t stream) {
    (void)in_sizes; (void)n_in; (void)out_size; (void)d_ws; (void)ws_size;
    const float* q    = (const float*)d_in[0];
    const float* k    = (const float*)d_in[1];
    const float* v    = (const float*)d_in[2];
    const float* g    = (const float*)d_in[3];
    const float* beta = (const float*)d_in[4];
    float* out = (float*)d_out;

    const size_t shmem = (size_t)(K_DIM * SROW          // state
                                  + 6 * CHUNK * LROW    // q/k/g ping-pong
                                  + 2 * CHUNK * VROW    // v ping-pong
                                  + CHUNK * LROW        // kbh
                                  + CHUNK * VROW        // U
                                  + 2 * 16 * 20         // As, Aq
                                  + K_DIM + CHUNK)      // lamC, betaS
                         * sizeof(float);               // ~110 KB
    (void)hipFuncSetAttribute((const void*)kda_fwd_kernel,
                              hipFuncAttributeMaxDynamicSharedMemorySize, (int)shmem);

    dim3 grid(B_DIM * H_DIM * VSPLIT);   // 128 workgroups
    dim3 block(NTHREADS);                // 4 wave32 waves
    kda_fwd_kernel<<<grid, block, shmem, stream>>>(q, k, v, g, beta, out);
}